// FlowWarping_76106820485747
// MI455X (gfx1250) — compile-verified
//
#include <hip/hip_runtime.h>

// FlowWarping (bilinear grid_sample with zero padding) for MI455X (gfx1250).
//
// Memory-bound op: 256 MiB compulsory HBM traffic -> ~11us floor at 23.3 TB/s.
// No matrix contraction exists, so WMMA is inapplicable; the gfx1250-specific
// wins are cache temporal hints (NT for the read-once flow / write-once out,
// regular-temporal for the ~4x-reused img so the 192MB L2 keeps it resident)
// and wave32-friendly 256-thread blocks with x-major lane mapping for
// coalescing. global_prefetch_b8 warms WGP$ for the two gather rows while the
// weight arithmetic executes.

#define BDIM 256

__global__ __launch_bounds__(BDIM) void flow_warp_bilinear_kernel(
    const float* __restrict__ img,   // [B,C,H,W]
    const float* __restrict__ flow,  // [B,2,H,W]
    float* __restrict__ out,         // [B,C,H,W]
    int B, int C, int H, int W)
{
    const int x = blockIdx.x * BDIM + threadIdx.x;
    const int y = blockIdx.y;
    const int b = blockIdx.z;
    if (x >= W) return;

    const long HW  = (long)H * (long)W;
    const long pix = (long)y * W + x;

    // ---- flow: read-once -> non-temporal loads (don't pollute L2) ----
    const float* fl = flow + (long)b * 2 * HW;
    const float dx = __builtin_nontemporal_load(fl + pix);
    const float dy = __builtin_nontemporal_load(fl + HW + pix);

    // ---- faithful normalize -> unnormalize round trip (align_corners) ----
    const float vx = (float)x + dx;
    const float vy = (float)y + dy;
    const float wm1 = (float)(W - 1);
    const float hm1 = (float)(H - 1);
    const float iw = 1.0f / fmaxf(wm1, 1.0f);
    const float ih = 1.0f / fmaxf(hm1, 1.0f);
    const float nx = 2.0f * vx * iw - 1.0f;
    const float ny = 2.0f * vy * ih - 1.0f;
    const float px = (nx + 1.0f) * 0.5f * wm1;
    const float py = (ny + 1.0f) * 0.5f * hm1;

    // ---- bilinear corners ----
    const float x0f = floorf(px);
    const float y0f = floorf(py);
    const int x0 = (int)x0f;
    const int y0 = (int)y0f;
    const int x1 = x0 + 1;
    const int y1 = y0 + 1;
    const float wx1 = px - x0f;
    const float wy1 = py - y0f;
    const float wx0 = 1.0f - wx1;
    const float wy0 = 1.0f - wy1;

    // Zero padding: fold validity into the weights (branchless, no divergence).
    const bool vx0 = (x0 >= 0) & (x0 <= W - 1);
    const bool vx1 = (x1 >= 0) & (x1 <= W - 1);
    const bool vy0 = (y0 >= 0) & (y0 <= H - 1);
    const bool vy1 = (y1 >= 0) & (y1 <= H - 1);

    const float w00 = (vy0 & vx0) ? (wy0 * wx0) : 0.0f;
    const float w01 = (vy0 & vx1) ? (wy0 * wx1) : 0.0f;
    const float w10 = (vy1 & vx0) ? (wy1 * wx0) : 0.0f;
    const float w11 = (vy1 & vx1) ? (wy1 * wx1) : 0.0f;

    const int x0c = min(max(x0, 0), W - 1);
    const int x1c = min(max(x1, 0), W - 1);
    const int y0c = min(max(y0, 0), H - 1);
    const int y1c = min(max(y1, 0), H - 1);

    const long o00 = (long)y0c * W + x0c;
    const long o01 = (long)y0c * W + x1c;
    const long o10 = (long)y1c * W + x0c;
    const long o11 = (long)y1c * W + x1c;

    const float* ip = img + (long)b * C * HW;
    float*       op = out + (long)b * C * HW;

    // Warm WGP$ for the two gather rows (gfx1250 global_prefetch_b8) while
    // the dependent weight math above retires.
    __builtin_prefetch(ip + o00, 0, 3);
    __builtin_prefetch(ip + o10, 0, 3);

    // img: ~4x reuse across neighboring pixels -> regular-temporal loads.
#pragma unroll
    for (int c = 0; c < 3; ++c) {
        const float v00 = ip[o00];
        const float v01 = ip[o01];
        const float v10 = ip[o10];
        const float v11 = ip[o11];
        const float r = v00 * w00 + v01 * w01 + v10 * w10 + v11 * w11;
        // out: write-once -> non-temporal store (keep L2 for img reuse).
        __builtin_nontemporal_store(r, op + pix);
        ip += HW;
        op += HW;
    }
}

extern "C" void kernel_launch(void* const* d_in, const int* in_sizes, int n_in,
                              void* d_out, int out_size, void* d_ws, size_t ws_size,
                              hipStream_t stream) {
    const float* img  = (const float*)d_in[0];
    const float* flow = (const float*)d_in[1];
    float* out = (float*)d_out;

    const int C = 3, H = 512, W = 512;
    const int B = in_sizes[0] / (C * H * W);  // 32 for the reference shapes

    dim3 block(BDIM, 1, 1);
    dim3 grid((W + BDIM - 1) / BDIM, H, B);
    flow_warp_bilinear_kernel<<<grid, block, 0, stream>>>(img, flow, out, B, C, H, W);
}